// SSMLayer_27754078667000
// MI455X (gfx1250) — compile-verified
//
#include <hip/hip_runtime.h>
#include <cstdint>
#include <cstddef>

// Pairwise L1 self-similarity matrix + global min-max normalization.
// M=4096, D=16. Memory-bound (67MB out @ 23.3TB/s ~ 3us floor); |a-b| is not
// bilinear so WMMA loses here (thermometer trick inflates K by 255x -> ~20us
// of iu8 WMMA vs ~5us of VALU). CDNA5-specific paths used:
//  - global_load_async_to_lds_b128 + s_wait_asynccnt (ASYNCcnt engine)
//  - packed f32 subtract expressed as native <2 x float> arithmetic so the
//    backend may select V_PK_ADF_F32-with-neg (VOP3P) WITHOUT inline-asm
//    register-allocator fences (round-2 asm version spilled: vmem 19->289).

#define MDIM 4096
#define DDIM 16
#define BI   128   // i-rows per block
#define BJ   64    // j-rows per block
#define NI   4     // i-rows per thread
#define NJ   8     // j-cols per thread
#define TPB  256   // 8 wave32 per block

typedef __attribute__((ext_vector_type(2))) float v2f;

__global__ __launch_bounds__(64)
void ssm_init_kernel(unsigned int* __restrict__ mm) {
    if (threadIdx.x == 0) {
        mm[0] = 0u;           // running max (uint bits of nonneg float)
        mm[1] = 0x7F800000u;  // running min (+inf bits)
    }
}

__global__ __launch_bounds__(TPB)
void ssm_dist_kernel(const float* __restrict__ x,
                     float* __restrict__ out,
                     unsigned int* __restrict__ mm) {
    __shared__ float4 si[BI][4];  // 128 rows * 16 f32 = 8 KB
    __shared__ float4 sj[BJ][4];  //  64 rows * 16 f32 = 4 KB

    const int tid = threadIdx.x;
    const int I0  = blockIdx.y * BI;
    const int J0  = blockIdx.x * BJ;

    // ---- CDNA5 async stage: x[I0:I0+BI) and x[J0:J0+BJ) -> LDS (ASYNCcnt) --
    {
        const unsigned siBase = (unsigned)(uintptr_t)&si[0][0];
        const unsigned sjBase = (unsigned)(uintptr_t)&sj[0][0];

        unsigned ldsA = siBase + (unsigned)tid * 16u;
        unsigned gA   = (unsigned)((I0 * DDIM) * 4 + tid * 16);
        asm volatile("global_load_async_to_lds_b128 %0, %1, %2"
                     :: "v"(ldsA), "v"(gA), "s"(x) : "memory");

        unsigned ldsA2 = siBase + (unsigned)(tid + TPB) * 16u;
        unsigned gA2   = (unsigned)((I0 * DDIM) * 4 + (tid + TPB) * 16);
        asm volatile("global_load_async_to_lds_b128 %0, %1, %2"
                     :: "v"(ldsA2), "v"(gA2), "s"(x) : "memory");

        unsigned ldsB = sjBase + (unsigned)tid * 16u;
        unsigned gB   = (unsigned)((J0 * DDIM) * 4 + tid * 16);
        asm volatile("global_load_async_to_lds_b128 %0, %1, %2"
                     :: "v"(ldsB), "v"(gB), "s"(x) : "memory");

        asm volatile("s_wait_asynccnt 0" ::: "memory");
    }
    __syncthreads();

    const int ti = tid >> 3;  // 0..31 -> i-group
    const int tj = tid & 7;   // 0..7  -> j-group

    // This thread's 4 i-rows, packed as 8x float2 each (64 VGPRs total)
    v2f a2[NI][DDIM / 2];
#pragma unroll
    for (int r = 0; r < NI; ++r) {
#pragma unroll
        for (int q = 0; q < 4; ++q) {
            float4 v = si[ti * NI + r][q];
            a2[r][q * 2 + 0] = (v2f){v.x, v.y};
            a2[r][q * 2 + 1] = (v2f){v.z, v.w};
        }
    }

    float acc[NI][NJ];
#pragma unroll
    for (int jj = 0; jj < NJ; ++jj) {
        const int jr = tj * NJ + jj;
        v2f b2[DDIM / 2];
#pragma unroll
        for (int q = 0; q < 4; ++q) {
            float4 v = sj[jr][q];  // ds_load_b128, broadcast-friendly
            b2[q * 2 + 0] = (v2f){v.x, v.y};
            b2[q * 2 + 1] = (v2f){v.z, v.w};
        }
#pragma unroll
        for (int r = 0; r < NI; ++r) {
            // 4 partial sums break the FP-add dependency chain;
            // abs folds into the v_add_f32 |src1| modifier.
            float s0 = 0.f, s1 = 0.f, s2 = 0.f, s3 = 0.f;
#pragma unroll
            for (int p = 0; p < DDIM / 2; p += 2) {
                v2f t0 = a2[r][p + 0] - b2[p + 0];  // packed sub (or 2x v_sub)
                v2f t1 = a2[r][p + 1] - b2[p + 1];
                s0 += __builtin_fabsf(t0.x);
                s1 += __builtin_fabsf(t0.y);
                s2 += __builtin_fabsf(t1.x);
                s3 += __builtin_fabsf(t1.y);
            }
            acc[r][jj] = (s0 + s1) + (s2 + s3);
        }
    }

    // Store (2x b128 per row) + local min/max
    float mx = 0.0f;
    float mn = __builtin_inff();
#pragma unroll
    for (int r = 0; r < NI; ++r) {
        const int row = I0 + ti * NI + r;
        float4* o = (float4*)(out + (size_t)row * MDIM + (size_t)(J0 + tj * NJ));
        float4 v0 = make_float4(acc[r][0], acc[r][1], acc[r][2], acc[r][3]);
        float4 v1 = make_float4(acc[r][4], acc[r][5], acc[r][6], acc[r][7]);
        o[0] = v0;
        o[1] = v1;
#pragma unroll
        for (int jj = 0; jj < NJ; ++jj) {
            mx = fmaxf(mx, acc[r][jj]);
            mn = fminf(mn, acc[r][jj]);
        }
    }

    // wave32 butterfly reduction, then one atomic per wave.
#pragma unroll
    for (int off = 16; off > 0; off >>= 1) {
        mx = fmaxf(mx, __shfl_xor(mx, off, 32));
        mn = fminf(mn, __shfl_xor(mn, off, 32));
    }
    if ((tid & 31) == 0) {
        // nonneg floats: uint-bit ordering == float ordering
        atomicMax(&mm[0], __float_as_uint(mx));
        atomicMin(&mm[1], __float_as_uint(mn));
    }
}

__global__ __launch_bounds__(TPB)
void ssm_norm_kernel(float* __restrict__ out,
                     const unsigned int* __restrict__ mm) {
    const float hi  = __uint_as_float(mm[0]);
    const float lo  = __uint_as_float(mm[1]);   // exactly 0 (diagonal)
    const float inv = 1.0f / (hi - lo);
    const size_t idx = (size_t)blockIdx.x * TPB + threadIdx.x;
    float4* o = (float4*)out;
    float4 v = o[idx];
    v.x = (v.x - lo) * inv;
    v.y = (v.y - lo) * inv;
    v.z = (v.z - lo) * inv;
    v.w = (v.w - lo) * inv;
    o[idx] = v;
}

extern "C" void kernel_launch(void* const* d_in, const int* in_sizes, int n_in,
                              void* d_out, int out_size, void* d_ws, size_t ws_size,
                              hipStream_t stream) {
    (void)in_sizes; (void)n_in; (void)ws_size;
    const float* x = (const float*)d_in[0];
    float* out = (float*)d_out;
    unsigned int* mm = (unsigned int*)d_ws;  // 8 bytes of scratch

    ssm_init_kernel<<<dim3(1), dim3(64), 0, stream>>>(mm);

    dim3 grid(MDIM / BJ, MDIM / BI);  // (64, 32) = 2048 blocks
    ssm_dist_kernel<<<grid, dim3(TPB), 0, stream>>>(x, out, mm);

    const int n4 = out_size / 4;  // 4,194,304 float4
    ssm_norm_kernel<<<dim3(n4 / TPB), dim3(TPB), 0, stream>>>(out, mm);
}